// CubicCatmullRomSpline_24489903522392
// MI455X (gfx1250) — compile-verified
//
#include <hip/hip_runtime.h>
#include <hip/hip_bf16.h>
#include <stdint.h>

typedef float v4f __attribute__((ext_vector_type(4)));

#define NGRID 43   // G+2 grid knots
#define NTAB  40   // packed bucket table entries, bc in [1,39]

// coefs = concat(co[0:21], {0}, co[21:42]) without materializing it
__device__ __forceinline__ float coef_at(const float* __restrict__ co, int i) {
    if (i == 21) return 0.0f;
    return (i < 21) ? co[i] : co[i - 1];
}

__device__ __forceinline__ float eval_one(float x,
                                          const float* __restrict__ sgrid,
                                          const v4f*  __restrict__ stab,
                                          float g8, float g16, float g24,
                                          float g32, float g40)
{
    // b = max{ i : grid[i] <= x }  (searchsorted(..., 'right') - 1, floored at 0)
    // Top 3 levels of the binary search compare against uniform SGPR pivots
    // (zero LDS traffic); bottom 3 levels read LDS (broadcast-friendly).
    int b = 0;
    if (g32 <= x) b = 32;
    if (b == 0 && g16 <= x) b = 16;
    float p8 = (b == 0) ? g8 : ((b == 16) ? g24 : g40);
    if (p8 <= x) b += 8;
#pragma unroll
    for (int s = 4; s > 0; s >>= 1) {
        int idx  = b + s;
        int cidx = (idx <= NGRID - 1) ? idx : (NGRID - 1);
        float v  = sgrid[cidx];                       // ds_load_b32
        if ((idx <= NGRID - 1) && (v <= x)) b = idx;
    }

    bool valid = (b >= 1) && (b <= NGRID - 4);
    int  bc    = b < 1 ? 1 : (b > NGRID - 4 ? NGRID - 4 : b);

    v4f e0 = stab[2 * bc];       // {g_lo, 1/dg, alpha, c0}  ds_load_b128
    v4f e1 = stab[2 * bc + 1];   // {c1, c2, c3, 0}          ds_load_b128

    float t  = (x - e0.x) * e0.y;
    float a  = e0.z;
    float t2 = t * t;
    float t3 = t2 * t;
    float t4 = t2 * t2;
    float f0 = 0.5f * (-t + 2.0f * (1.0f + a) * t2 - (1.0f + 4.0f * a) * t3 + 2.0f * a * t4);
    float f1 = 0.5f * ( 2.0f - (5.0f + 2.0f * a) * t2 + (3.0f + 4.0f * a) * t3 - 2.0f * a * t4);
    float f2 = 0.5f * ( t + 2.0f * (2.0f - a) * t2 - (3.0f - 4.0f * a) * t3 - 2.0f * a * t4);
    float f3 = 0.5f * (-(1.0f - 2.0f * a) * t2 + (1.0f - 4.0f * a) * t3 + 2.0f * a * t4);

    float y = f0 * e0.w + f1 * e1.x + f2 * e1.y + f3 * e1.z;
    return valid ? y : 0.0f;
}

__global__ void __launch_bounds__(256)
CubicCatmullRomSpline_24489903522392_kernel(const float* __restrict__ x,
                                            const float* __restrict__ co,
                                            const float* __restrict__ alphas,
                                            const float* __restrict__ grid,
                                            float* __restrict__ out,
                                            int n)
{
    __shared__ float sgrid[NGRID];
    __shared__ v4f   stab[2 * NTAB];

    const int tid = threadIdx.x;

    // Uniform SGPR pivots for the top of the search tree (s_load, shared by wave).
    const float g8  = grid[8];
    const float g16 = grid[16];
    const float g24 = grid[24];
    const float g32 = grid[32];
    const float g40 = grid[40];

    // --- Stage grid knots into LDS via gfx1250 async global->LDS copy ---
    if (tid < NGRID) {
        uint32_t ldsOff = (uint32_t)(uintptr_t)(&sgrid[tid]);  // low 32b of generic addr = LDS offset
        uint32_t gOff   = (uint32_t)(tid * 4u);
        asm volatile("global_load_async_to_lds_b32 %0, %1, %2"
                     :
                     : "v"(ldsOff), "v"(gOff),
                       "s"((uint64_t)(uintptr_t)grid)
                     : "memory");
    }

    // --- Build packed per-bucket table: {g_lo, 1/dg, alpha, c0 | c1, c2, c3, 0} ---
    // blockDim.x (256) > NTAB (40): flat predicated build, no loop.
    if (tid < NTAB) {
        const int k = tid;
        v4f e0 = {0.f, 0.f, 0.f, 0.f};
        v4f e1 = {0.f, 0.f, 0.f, 0.f};
        if (k >= 1 && k <= NGRID - 4) {
            float glo = grid[k];
            float ghi = grid[k + 1];
            float a   = alphas[k - 1];
            e0 = (v4f){glo, 1.0f / (ghi - glo), a, coef_at(co, k - 1)};
            e1 = (v4f){coef_at(co, k), coef_at(co, k + 1), coef_at(co, k + 2), 0.f};
        }
        stab[2 * k]     = e0;
        stab[2 * k + 1] = e1;
    }

    // Complete the async LDS fill, then make it visible to all waves.
    asm volatile("s_wait_asynccnt 0x0" ::: "memory");
    __syncthreads();

    const int n4     = n >> 2;
    const int stride = gridDim.x * blockDim.x;
    const int gtid   = blockIdx.x * blockDim.x + tid;

    const v4f* __restrict__ x4 = (const v4f*)x;
    v4f* __restrict__       o4 = (v4f*)out;

    // Streaming main loop: 16B NT loads/stores (global_load_b128 / global_store_b128, TH=NT)
    // so the 268 MB stream bypasses persistent L2 residency.
    for (int i = gtid; i < n4; i += stride) {
        v4f xv = __builtin_nontemporal_load(&x4[i]);
        v4f yv;
        yv.x = eval_one(xv.x, sgrid, stab, g8, g16, g24, g32, g40);
        yv.y = eval_one(xv.y, sgrid, stab, g8, g16, g24, g32, g40);
        yv.z = eval_one(xv.z, sgrid, stab, g8, g16, g24, g32, g40);
        yv.w = eval_one(xv.w, sgrid, stab, g8, g16, g24, g32, g40);
        __builtin_nontemporal_store(yv, &o4[i]);
    }

    // Scalar tail (n not divisible by 4)
    for (int i = (n4 << 2) + gtid; i < n; i += stride) {
        float xv = __builtin_nontemporal_load(&x[i]);
        float yv = eval_one(xv, sgrid, stab, g8, g16, g24, g32, g40);
        __builtin_nontemporal_store(yv, &out[i]);
    }
}

extern "C" void kernel_launch(void* const* d_in, const int* in_sizes, int n_in,
                              void* d_out, int out_size, void* d_ws, size_t ws_size,
                              hipStream_t stream) {
    (void)in_sizes; (void)n_in; (void)d_ws; (void)ws_size;
    const float* x      = (const float*)d_in[0];
    const float* co     = (const float*)d_in[1];
    const float* alphas = (const float*)d_in[2];
    const float* grid   = (const float*)d_in[3];
    float*       out    = (float*)d_out;

    const int n       = out_size;          // 33,554,432
    const int threads = 256;               // 8 wave32s per block
    int n4 = n >> 2;
    int blocks = (n4 + threads - 1) / threads;
    if (blocks > 4096) blocks = 4096;      // grid-stride; amortizes LDS setup per block
    if (blocks < 1) blocks = 1;

    CubicCatmullRomSpline_24489903522392_kernel<<<blocks, threads, 0, stream>>>(
        x, co, alphas, grid, out, n);
}